// PCNLayer_8564164788725
// MI455X (gfx1250) — compile-verified
//
#include <hip/hip_runtime.h>
#include <math.h>

// Problem constants (from reference)
#define NN 5000
#define BB 4
#define HH 16
#define CC 64
#define SS 64
#define RR 8
#define TT 16
#define OO 64
#define WOUT 14          // T - (K-1)*DIL = 16 - 2
#define WTOT 28          // fwd + bwd
#define YROW 68          // padded LDS row stride (floats) per (t,h): 64 data + 4 pad
#define YT   (HH*YROW)   // floats per t-slice in LDS (1088)
#define WROW 132         // padded LDS row stride for weight matrix (128 data + 4 pad)

typedef float v2f __attribute__((ext_vector_type(2)));
typedef float v8f __attribute__((ext_vector_type(8)));

// GEMM view per (s,r,b):
//   D[row, col] = sum_ck Wmat[row, ck] * Yp[ck, col]
//   rows 0..63  = w1 out-channels (linear), rows 64..127 = w2 (gate)
//   ck = kblock*64 + c  (kblock = conv tap 0/1, c = input channel)
//   col tile ct: 0..13 fwd (t = w + 2*kb), 14..27 bwd (t = 15 - w - 2*kb)
__global__ __launch_bounds__(256, 1)
void pcn_glu_wmma(const float* __restrict__ x,
                  const float* __restrict__ w1,
                  const float* __restrict__ b1,
                  const float* __restrict__ w2,
                  const float* __restrict__ b2,
                  const int*   __restrict__ tidx,
                  float* __restrict__ out)
{
    __shared__ float ylds[TT * YT];    // 69,632 B, conflict-free padded
    __shared__ float wlds[128 * WROW]; // 67,584 B, stacked w1|w2 in GEMM layout
    __shared__ int   nodes[RR * TT];   // trajectory node ids for this s

    const int tid  = threadIdx.x;
    const int wave = tid >> 5;         // 0..7
    const int lane = tid & 31;
    const int lh   = lane & 15;        // row/col within 16-wide fragment
    const int hi   = lane >> 4;        // half-wave select (K-pair / M-half)
    const int p    = wave & 3;         // row-pair tile: o in [p*16, p*16+16)
    const int cg   = wave >> 2;        // column group (14 col tiles each)

    const int bid = blockIdx.x;        // 256 = S*B
    const int s = bid >> 2;
    const int b = bid & 3;

    if (tid < RR * TT) nodes[tid] = tidx[s * RR * TT + tid];

    // ---- Stage weights once: wlds[row*WROW + ck], row<64 => w1, row>=64 => w2
    // w[o][c][0][k] flat = o*128 + c*2 + k ; ck = kb*64 + c
    for (int i = tid; i < 128 * 128; i += 256) {
        const int row = i >> 7;
        const int ck  = i & 127;
        const int src = ((ck & 63) * 2) + (ck >> 6);
        const float v = (row < 64) ? w1[row * (CC * 2) + src]
                                   : w2[(row - 64) * (CC * 2) + src];
        wlds[row * WROW + ck] = v;
    }
    __syncthreads();   // wlds ready for A-fragment register loads

    // ---- Hoist all A fragments into registers (reused 14 cols x 8 r) ----
    // A 16x4 f32 layout: lanes 0-15 hold M=0..15 with {K0,K1}; lanes 16-31 {K2,K3}.
    const int alin_base  = (p * 16 + lh) * WROW + hi * 2;
    const int agate_base = alin_base + 64 * WROW;
    v2f Alin[32], Agate[32];
    #pragma unroll
    for (int ks = 0; ks < 32; ++ks) {
        Alin[ks]  = *(const v2f*)&wlds[alin_base  + ks * 4];
        Agate[ks] = *(const v2f*)&wlds[agate_base + ks * 4];
    }

    // Biases for this lane's 8 output rows (C/D layout: VGPR v -> M = v + hi*8)
    float bias1[8], bias2[8];
    #pragma unroll
    for (int v = 0; v < 8; ++v) {
        bias1[v] = b1[p * 16 + hi * 8 + v];
        bias2[v] = b2[p * 16 + hi * 8 + v];
    }

    const int yb_base = lh * YROW + hi * 2;

    const v8f vzero = {};
    v8f res[14];
    #pragma unroll
    for (int j = 0; j < 14; ++j) res[j] = vzero;

    for (int r = 0; r < RR; ++r) {
        __syncthreads();   // prior reads of ylds done; also covers nodes fill
        // ---- Stage Y for this (s,r,b): wave w copies t-slices w and w+8 ----
        #pragma unroll
        for (int half = 0; half < 2; ++half) {
            const int t = wave + half * 8;
            const int node = nodes[r * TT + t];
            const float* src = x + ((size_t)node * BB + b) * (HH * CC);
            #pragma unroll
            for (int i = 0; i < 8; ++i) {
                const int e  = i * 32 + lane;   // 0..255 float4's
                const int h  = e >> 4;
                const int c4 = e & 15;
                const float4 vv = *(const float4*)(src + h * CC + c4 * 4);
                *(float4*)&ylds[t * YT + h * YROW + c4 * 4] = vv;
            }
        }
        __syncthreads();

        // ---- GEMM + GLU for this wave's 14 column tiles ----
        #pragma unroll
        for (int j = 0; j < 14; ++j) {
            const int ct  = cg * 14 + j;        // 0..27 global column tile
            const int dir = (ct >= WOUT) ? 1 : 0;
            const int wl  = dir ? (ct - WOUT) : ct;

            // Burst-load all 32 B fragments for this tile into registers
            // (merges to 16 ds_load_2addr_b64); pays LDS latency once per
            // 64 wmma instead of once per 4.
            v2f Bbuf[32];
            #pragma unroll
            for (int ks = 0; ks < 32; ++ks) {
                const int kb = ks >> 4;
                const int t  = dir ? (TT - 1 - wl - 2 * kb) : (wl + 2 * kb);
                const int cb = (ks * 4) & 63;
                // B 4x16 f32: lanes 0-15 cols 0..15 {K0,K1}; lanes 16-31 {K2,K3}
                Bbuf[ks] = *(const v2f*)&ylds[t * YT + yb_base + cb];
            }

            v8f Cl = vzero, Cg = vzero;
            #pragma unroll
            for (int ks = 0; ks < 32; ++ks) {
                Cl = __builtin_amdgcn_wmma_f32_16x16x4_f32(
                        false, Alin[ks], false, Bbuf[ks], (short)0, Cl, false, false);
                Cg = __builtin_amdgcn_wmma_f32_16x16x4_f32(
                        false, Agate[ks], false, Bbuf[ks], (short)0, Cg, false, false);
            }
            #pragma unroll
            for (int v = 0; v < 8; ++v) {
                const float lin = Cl[v] + bias1[v];
                const float g   = Cg[v] + bias2[v];
                res[j][v] += lin * (1.0f / (1.0f + __expf(-g)));   // GLU
            }
        }
    }

    // ---- Store mean over R: out[b][s][o][h][wtot], scale by 1/R ----
    #pragma unroll
    for (int j = 0; j < 14; ++j) {
        const int ct = cg * 14 + j;
        #pragma unroll
        for (int v = 0; v < 8; ++v) {
            const int oo = p * 16 + hi * 8 + v;
            out[((((size_t)b * SS + s) * OO + oo) * HH + lh) * WTOT + ct] =
                res[j][v] * 0.125f;
        }
    }
}

extern "C" void kernel_launch(void* const* d_in, const int* in_sizes, int n_in,
                              void* d_out, int out_size, void* d_ws, size_t ws_size,
                              hipStream_t stream) {
    (void)in_sizes; (void)n_in; (void)out_size; (void)d_ws; (void)ws_size;
    const float* x  = (const float*)d_in[0];
    const float* w1 = (const float*)d_in[1];
    const float* b1 = (const float*)d_in[2];
    const float* w2 = (const float*)d_in[3];
    const float* b2 = (const float*)d_in[4];
    const int*   ti = (const int*)d_in[5];   // JAX default x64-disabled -> int32
    float* out = (float*)d_out;

    dim3 grid(SS * BB);   // 256 workgroups: one (s,b) each
    dim3 block(256);      // 8 wave32 waves, large per-wave VGPR budget
    hipLaunchKernelGGL(pcn_glu_wmma, grid, block, 0, stream,
                       x, w1, b1, w2, b2, ti, out);
}